// EvolutionModel_53695681135134
// MI455X (gfx1250) — compile-verified
//
#include <hip/hip_runtime.h>
#include <math.h>

// ---------------------------------------------------------------------------
// EvolutionModel for gfx1250 (CDNA5).  Hot loop (point-in-tetrahedron search
// for 16384 rays x 2000 tetra, 8 times) is mapped to V_WMMA_F32_16X16X4_F32.
// ---------------------------------------------------------------------------

#define TWO_PI_F 6.28318530717958647692f

typedef float v2f __attribute__((ext_vector_type(2)));
typedef float v8f __attribute__((ext_vector_type(8)));

struct V3 { float x, y, z; };

__device__ __forceinline__ V3 vsub(V3 a, V3 b) { return {a.x - b.x, a.y - b.y, a.z - b.z}; }
__device__ __forceinline__ float dot3(V3 a, V3 b) { return a.x * b.x + a.y * b.y + a.z * b.z; }
__device__ __forceinline__ V3 cross3(V3 a, V3 b) {
    return {a.y * b.z - a.z * b.y, a.z * b.x - a.x * b.z, a.x * b.y - a.y * b.x};
}
__device__ __forceinline__ int imodp(int a, int T) { int r = a % T; return (r < 0) ? r + T : r; }
__device__ __forceinline__ float wrap2pi(float p) {
    float r = fmodf(p, TWO_PI_F);
    if (r < 0.0f) r += TWO_PI_F;
    return r;   // NaN passes through (NaN < 0 is false)
}

// Standard adjugate 4x4 inverse (matches jnp.linalg.inv up to fp noise).
__device__ void inv4x4(const float m[16], float inv[16]) {
    inv[0]  =  m[5]*m[10]*m[15] - m[5]*m[11]*m[14] - m[9]*m[6]*m[15] + m[9]*m[7]*m[14] + m[13]*m[6]*m[11] - m[13]*m[7]*m[10];
    inv[4]  = -m[4]*m[10]*m[15] + m[4]*m[11]*m[14] + m[8]*m[6]*m[15] - m[8]*m[7]*m[14] - m[12]*m[6]*m[11] + m[12]*m[7]*m[10];
    inv[8]  =  m[4]*m[9]*m[15]  - m[4]*m[11]*m[13] - m[8]*m[5]*m[15] + m[8]*m[7]*m[13] + m[12]*m[5]*m[11] - m[12]*m[7]*m[9];
    inv[12] = -m[4]*m[9]*m[14]  + m[4]*m[10]*m[13] + m[8]*m[5]*m[14] - m[8]*m[6]*m[13] - m[12]*m[5]*m[10] + m[12]*m[6]*m[9];
    inv[1]  = -m[1]*m[10]*m[15] + m[1]*m[11]*m[14] + m[9]*m[2]*m[15] - m[9]*m[3]*m[14] - m[13]*m[2]*m[11] + m[13]*m[3]*m[10];
    inv[5]  =  m[0]*m[10]*m[15] - m[0]*m[11]*m[14] - m[8]*m[2]*m[15] + m[8]*m[3]*m[14] + m[12]*m[2]*m[11] - m[12]*m[3]*m[10];
    inv[9]  = -m[0]*m[9]*m[15]  + m[0]*m[11]*m[13] + m[8]*m[1]*m[15] - m[8]*m[3]*m[13] - m[12]*m[1]*m[11] + m[12]*m[3]*m[9];
    inv[13] =  m[0]*m[9]*m[14]  - m[0]*m[10]*m[13] - m[8]*m[1]*m[14] + m[8]*m[2]*m[13] + m[12]*m[1]*m[10] - m[12]*m[2]*m[9];
    inv[2]  =  m[1]*m[6]*m[15]  - m[1]*m[7]*m[14]  - m[5]*m[2]*m[15] + m[5]*m[3]*m[14] + m[13]*m[2]*m[7]  - m[13]*m[3]*m[6];
    inv[6]  = -m[0]*m[6]*m[15]  + m[0]*m[7]*m[14]  + m[4]*m[2]*m[15] - m[4]*m[3]*m[14] - m[12]*m[2]*m[7]  + m[12]*m[3]*m[6];
    inv[10] =  m[0]*m[5]*m[15]  - m[0]*m[7]*m[13]  - m[4]*m[1]*m[15] + m[4]*m[3]*m[13] + m[12]*m[1]*m[7]  - m[12]*m[3]*m[5];
    inv[14] = -m[0]*m[5]*m[14]  + m[0]*m[6]*m[13]  + m[4]*m[1]*m[14] - m[4]*m[2]*m[13] - m[12]*m[1]*m[6]  + m[12]*m[2]*m[5];
    inv[3]  = -m[1]*m[6]*m[11]  + m[1]*m[7]*m[10]  + m[5]*m[2]*m[11] - m[5]*m[3]*m[10] - m[9]*m[2]*m[7]   + m[9]*m[3]*m[6];
    inv[7]  =  m[0]*m[6]*m[11]  - m[0]*m[7]*m[10]  - m[4]*m[2]*m[11] + m[4]*m[3]*m[10] + m[8]*m[2]*m[7]   - m[8]*m[3]*m[6];
    inv[11] = -m[0]*m[5]*m[11]  + m[0]*m[7]*m[9]   + m[4]*m[1]*m[11] - m[4]*m[3]*m[9]  - m[8]*m[1]*m[7]   + m[8]*m[3]*m[5];
    inv[15] =  m[0]*m[5]*m[10]  - m[0]*m[6]*m[9]   - m[4]*m[1]*m[10] + m[4]*m[2]*m[9]  + m[8]*m[1]*m[6]   - m[8]*m[2]*m[5];
    float det = m[0]*inv[0] + m[1]*inv[4] + m[2]*inv[8] + m[3]*inv[12];
    float rd = 1.0f / det;
    for (int i = 0; i < 16; ++i) inv[i] *= rd;
}

// ---------------------------------------------------------------------------
// Per-tetra precompute: (a/||b||, n) for the optics, and Aext rows (4 per
// tetra) so that point-in-tetra == all 4 rows of Aext @ [p;1] >= 0.
// ---------------------------------------------------------------------------
__global__ void precompute_kernel(const float* __restrict__ pos,
                                  const int* __restrict__ tetra,   // (4,T)
                                  const float* __restrict__ nidx,
                                  const float* __restrict__ origin, // (T,3)
                                  const float* __restrict__ ort,    // (3,3,T)
                                  int T,
                                  float* __restrict__ Aext,         // (4T,4)
                                  float* __restrict__ aob,          // (T)
                                  float* __restrict__ nvec)         // (T,3)
{
    int t = blockIdx.x * blockDim.x + threadIdx.x;
    if (t >= T) return;

    int vid[4];
    for (int v = 0; v < 4; ++v) vid[v] = tetra[v * T + t];

    float M[16];
    for (int c = 0; c < 4; ++c) M[c] = 1.0f;
    for (int r = 0; r < 3; ++r)
        for (int c = 0; c < 4; ++c)
            M[(r + 1) * 4 + c] = pos[vid[c] * 3 + r];

    float K[16];
    inv4x4(M, K);

    float vv[4];
    for (int i = 0; i < 4; ++i) vv[i] = nidx[vid[i]];
    float ab[4];
    for (int j = 0; j < 4; ++j) {
        float s = 0.0f;
        for (int i = 0; i < 4; ++i) s += K[i * 4 + j] * vv[i];
        ab[j] = s;
    }
    float bx = ab[1], by = ab[2], bz = ab[3];
    float bn = sqrtf(bx * bx + by * by + bz * bz);
    aob[t] = ab[0] / bn;
    nvec[t * 3 + 0] = bx / bn;
    nvec[t * 3 + 1] = by / bn;
    nvec[t * 3 + 2] = bz / bn;

    float o0 = origin[t * 3 + 0], o1 = origin[t * 3 + 1], o2 = origin[t * 3 + 2];
    float s0 = 0.0f, s1 = 0.0f, s2 = 0.0f, sc = 0.0f;
    for (int i = 0; i < 3; ++i) {
        float mi0 = ort[(i * 3 + 0) * T + t];
        float mi1 = ort[(i * 3 + 1) * T + t];
        float mi2 = ort[(i * 3 + 2) * T + t];
        float ci = mi0 * o0 + mi1 * o1 + mi2 * o2;
        float* row = Aext + (size_t)(4 * t + i) * 4;
        row[0] = mi0; row[1] = mi1; row[2] = mi2; row[3] = -ci;
        s0 += mi0; s1 += mi1; s2 += mi2; sc += ci;
    }
    float* row3 = Aext + (size_t)(4 * t + 3) * 4;
    row3[0] = -s0; row3[1] = -s1; row3[2] = -s2; row3[3] = 1.0f + sc;  // 1 - sum(newp)
}

// ---------------------------------------------------------------------------
// init: r_hist slab0 = r0, m_buf = m0, d_hist slab0 = 0
// ---------------------------------------------------------------------------
__global__ void init_kernel(const float* __restrict__ r0, const float* __restrict__ m0,
                            float* __restrict__ rhist0, float* __restrict__ mbuf,
                            float* __restrict__ dhist0, int B)
{
    int i = blockIdx.x * blockDim.x + threadIdx.x;
    if (i < B * 3) { rhist0[i] = r0[i]; mbuf[i] = m0[i]; }
    if (i < B) dhist0[i] = 0.0f;
}

// ---------------------------------------------------------------------------
// WMMA point-in-tetra search.  One block (8 waves) handles 16 rays; the 8
// waves stride across the 4T/16 A-tiles (each tile = 4 tetrahedra), then the
// per-wave "last valid tetra" results are max-reduced through LDS.
// D layout: lane(n)<16 holds rows 0..7 (tetras 4k,4k+1) of ray n,
// lane(n+16) holds rows 8..15 (tetras 4k+2,4k+3).
// A-tile loads are software-pipelined; the wave id is readfirstlane'd so the
// k-loop is scalar (s_cbranch, no EXEC masking), and the valid-tetra select
// is two independent cndmax candidates so nothing gets sunk behind saveexec.
// ---------------------------------------------------------------------------
__global__ void __launch_bounds__(256)
find_tetra_kernel(const float* __restrict__ P,     // (B,3)
                  const float* __restrict__ Aext,  // (4T,4)
                  int* __restrict__ outIdx, int T, int B)
{
    const int NW = 8;                       // waves per block
    int wu   = __builtin_amdgcn_readfirstlane(threadIdx.x >> 5);  // scalar wave id
    int lane = threadIdx.x & 31;
    int half = lane >> 4;                   // 0 or 1
    int col  = lane & 15;                   // ray within tile / N column
    int rayBase = blockIdx.x * 16;
    int ray  = rayBase + col;
    int rr   = (ray < B) ? ray : (B - 1);   // keep all lanes live for WMMA

    float px = P[rr * 3 + 0];
    float py = P[rr * 3 + 1];
    float pz = P[rr * 3 + 2];
    v2f bop;                                // B 4x16: V0={K0|K2}, V1={K1|K3}
    bop.x = half ? pz : px;
    bop.y = half ? 1.0f : py;

    const int ntiles = (4 * T) >> 4;
    // per-lane A source: row = col, columns 2*half..2*half+1 of each tile
    const float* Abase = Aext + (size_t)col * 4 + 2 * half;

    int best = -1;
    if (wu < ntiles) {                      // scalar guard
        v2f a_cur = *(const v2f*)(Abase + (size_t)wu * 64);
        for (int k = wu; k < ntiles; k += NW) {
            int kn = k + NW;
            kn = (kn < ntiles) ? kn : k;    // clamp: harmless re-load on last iter
            v2f a_next = *(const v2f*)(Abase + (size_t)kn * 64);

            v8f acc = {0.f, 0.f, 0.f, 0.f, 0.f, 0.f, 0.f, 0.f};
            acc = __builtin_amdgcn_wmma_f32_16x16x4_f32(false, a_cur, false, bop,
                                                        (short)0, acc, false, false);
            // NaN >= 0 is false, so NaN rays never match (mirrors reference).
            int v0 = (int)(acc[0] >= 0.f) & (int)(acc[1] >= 0.f) &
                     (int)(acc[2] >= 0.f) & (int)(acc[3] >= 0.f);
            int v1 = (int)(acc[4] >= 0.f) & (int)(acc[5] >= 0.f) &
                     (int)(acc[6] >= 0.f) & (int)(acc[7] >= 0.f);
            int base = 4 * k + 2 * half;
            int c0 = v0 ? base : -1;        // both candidates always computed:
            int c1 = v1 ? (base + 1) : -1;  // nothing to sink behind saveexec
            int cand = (c1 > c0) ? c1 : c0;
            best = (cand > best) ? cand : best;   // ascending k => max == last valid
            a_cur = a_next;
        }
    }
    int other = __shfl_xor(best, 16, 32);
    best = (best > other) ? best : other;

    __shared__ int sbest[NW][16];
    if (half == 0) sbest[wu][col] = best;
    __syncthreads();
    if (threadIdx.x < 16) {
        int m = sbest[0][threadIdx.x];
        for (int i = 1; i < NW; ++i) {
            int v = sbest[i][threadIdx.x];
            m = (v > m) ? v : m;
        }
        if (rayBase + threadIdx.x < B) outIdx[rayBase + threadIdx.x] = m;
    }
}

// ---------------------------------------------------------------------------
// do_intersection: angles where circular arc hits a face plane
// ---------------------------------------------------------------------------
__device__ void do_intersection(int face, V3 R, float Rn, V3 m, V3 rc,
                                const float* __restrict__ pos,
                                const int* __restrict__ fvert,
                                float* p1out, float* p2out)
{
    int i0 = fvert[face * 3 + 0], i1 = fvert[face * 3 + 1], i2 = fvert[face * 3 + 2];
    V3 vi = {pos[i0 * 3], pos[i0 * 3 + 1], pos[i0 * 3 + 2]};
    V3 vj = {pos[i1 * 3], pos[i1 * 3 + 1], pos[i1 * 3 + 2]};
    V3 vk = {pos[i2 * 3], pos[i2 * 3 + 1], pos[i2 * 3 + 2]};
    V3 ML = cross3(vsub(vj, vi), vsub(vk, vi));
    float QL = -dot3(vi, ML);
    float c1 = -dot3(ML, R);
    float c2 = Rn * dot3(ML, m);
    float c3 = dot3(ML, rc) + QL;
    float phi_eq = -2.0f * atanf(c1 / c2);
    float disc = c1 * c1 + c2 * c2 - c3 * c3;
    float s = (disc >= 0.0f) ? sqrtf(disc) : __builtin_nanf("");
    float den = c1 - c3;
    float phi1 = 2.0f * atanf((c2 + s) / den);
    float phi2 = 2.0f * atanf((c2 - s) / den);
    bool cond = (phi2 == 0.0f) && (c1 * c3 > 0.0f);
    if (cond) { phi1 = phi_eq; phi2 = phi_eq; }
    *p1out = wrap2pi(phi1);
    *p2out = wrap2pi(phi2);
}

// ---------------------------------------------------------------------------
// _evolve_in_tetra: one thread per ray
// ---------------------------------------------------------------------------
__global__ void evolve_kernel(const int* __restrict__ tidx,
                              const float* __restrict__ rin,   // r_hist slab s
                              float* __restrict__ rout,        // r_hist slab s+1
                              float* __restrict__ mbuf,        // in/out
                              float* __restrict__ dout,        // d_hist slab s+1
                              const float* __restrict__ aob,
                              const float* __restrict__ nvec,
                              const float* __restrict__ pos,
                              const int* __restrict__ tface,   // (T,4)
                              const int* __restrict__ fvert,   // (F,3)
                              const int* __restrict__ ftet,    // (F,2)
                              int T, int B)
{
    int b = blockIdx.x * blockDim.x + threadIdx.x;
    if (b >= B) return;

    int ti = tidx[b];
    int gi = imodp(ti, T);
    V3 rp = {rin[b * 3], rin[b * 3 + 1], rin[b * 3 + 2]};
    V3 m  = {mbuf[b * 3], mbuf[b * 3 + 1], mbuf[b * 3 + 2]};
    float a_ob = aob[gi];
    V3 n = {nvec[gi * 3], nvec[gi * 3 + 1], nvec[gi * 3 + 2]};

    V3 mn = cross3(m, n);
    float mnl = sqrtf(dot3(mn, mn));
    V3 q = {mn.x / mnl, mn.y / mnl, mn.z / mnl};
    V3 nq = cross3(n, q);
    float mndot = dot3(m, n);
    float dmnq = dot3(m, nq);
    float f = mndot / dmnq;
    float scal = dot3(rp, n) + a_ob;
    V3 vec = {n.x - f * nq.x, n.y - f * nq.y, n.z - f * nq.z};
    V3 rc = {rp.x - scal * vec.x, rp.y - scal * vec.y, rp.z - scal * vec.z};
    V3 R = vsub(rc, rp);
    float Rn = sqrtf(dot3(R, R));

    // ---- find intersection face among the tetra's 4 faces ----
    int faces[4];
    float phiR[4];
    for (int ff = 0; ff < 4; ++ff) {
        faces[ff] = tface[gi * 4 + ff];
        float p1, p2;
        do_intersection(faces[ff], R, Rn, m, rc, pos, fvert, &p1, &p2);
        // jnp.min (NaN-propagating) then nan_to_num(nan=10)
        phiR[ff] = (isnan(p1) || isnan(p2)) ? 10.0f : fminf(p1, p2);
    }
    int idx = 0; float phiE = phiR[0];
    for (int ff = 1; ff < 4; ++ff)
        if (phiR[ff] < phiE) { phiE = phiR[ff]; idx = ff; }   // first-min == jnp.argmin
    int hit_face = faces[idx];

    float cE = cosf(phiE), sE = sinf(phiE);
    V3 re = {rc.x - cE * R.x + Rn * sE * m.x,
             rc.y - cE * R.y + Rn * sE * m.y,
             rc.z - cE * R.z + Rn * sE * m.z};
    float isR = sE / Rn;
    V3 me = {cE * m.x + isR * R.x, cE * m.y + isR * R.y, cE * m.z + isR * R.z};

    int ft0 = ftet[hit_face * 2 + 0];
    int ft1 = ftet[hit_face * 2 + 1];
    int nt = (ft0 == ti) ? ft1 : ft0;

    // ---- straight-ray advance into the next tetra ----
    int gi2 = imodp(nt, T);
    float t_res = 0.0f;
    float min_t = 0.0f;
    for (int ff = 0; ff < 4; ++ff) {
        int face2 = tface[gi2 * 4 + ff];
        int i0 = fvert[face2 * 3 + 0], i1 = fvert[face2 * 3 + 1], i2 = fvert[face2 * 3 + 2];
        V3 vi = {pos[i0 * 3], pos[i0 * 3 + 1], pos[i0 * 3 + 2]};
        V3 vj = {pos[i1 * 3], pos[i1 * 3 + 1], pos[i1 * 3 + 2]};
        V3 vk = {pos[i2 * 3], pos[i2 * 3 + 1], pos[i2 * 3 + 2]};
        V3 Nf = cross3(vsub(vj, vi), vsub(vk, vi));
        float D = dot3(Nf, vi);
        float tf = (-dot3(Nf, re) + D) / dot3(Nf, me);
        bool mr = (face2 == hit_face);
        if (mr) t_res = tf;                       // last match wins (reference loop)
        float ta = ((tf > 0.0f) && !mr) ? tf : 1.0f;   // NaN>0 false -> 1.0
        min_t = (ff == 0) ? ta : fminf(min_t, ta);
    }
    float cand1 = (min_t - t_res) / 3.0f;
    float cand2 = t_res * 1.05f;
    float tnext = (isnan(cand1) || isnan(cand2)) ? __builtin_nanf("") : fmaxf(cand1, cand2);

    V3 re2 = {re.x + tnext * me.x, re.y + tnext * me.y, re.z + tnext * me.z};
    V3 dr = vsub(rp, re2);
    float dist = sqrtf(dot3(dr, dr));

    rout[b * 3] = re2.x; rout[b * 3 + 1] = re2.y; rout[b * 3 + 2] = re2.z;
    mbuf[b * 3] = me.x;  mbuf[b * 3 + 1] = me.y;  mbuf[b * 3 + 2] = me.z;
    dout[b] = dist;
}

// ---------------------------------------------------------------------------
// Final sampling.  One block (= SAMP threads) per ray; history staged in LDS.
// ---------------------------------------------------------------------------
__global__ void finalize_kernel(const float* __restrict__ zv,    // (B,SAMP)
                                const float* __restrict__ rhist, // 9 slabs (B,3)
                                const float* __restrict__ dhist, // 9 slabs (B)
                                float* __restrict__ out, int B, int SAMP)
{
    const int NH = 9;
    int b = blockIdx.x;
    int t = threadIdx.x;
    __shared__ float sd[NH];
    __shared__ float sr[NH][3];
    if (t == 0) {
        float run = 0.0f;
        for (int s = 0; s < NH; ++s) { run += dhist[(size_t)s * B + b]; sd[s] = run; }
    }
    if (t < NH * 3) {
        int s = t / 3, c = t % 3;
        sr[s][c] = rhist[((size_t)s * B + b) * 3 + c];
    }
    __syncthreads();
    if (t >= SAMP) return;

    float z = zv[(size_t)b * SAMP + t];

    int ip = 0, in2 = 0;
    float bp = 0.0f, bn2 = 0.0f;
    bool lockp = false, lockn = false, firstp = true, firstn = true;
    for (int k = 0; k < NH; ++k) {
        float tmd = z - sd[k];
        float pv = (tmd < 0.0f) ? 10.0f : tmd;   // NaN cond false -> keeps NaN
        float nv = (tmd > 0.0f) ? -10.0f : tmd;
        if (!lockp) {
            if (isnan(pv)) { ip = k; bp = pv; lockp = true; }        // np argmin: first NaN
            else if (firstp || pv < bp) { ip = k; bp = pv; firstp = false; }
        }
        if (!lockn) {
            if (isnan(nv)) { in2 = k; bn2 = nv; lockn = true; }      // np argmax: first NaN
            else if (firstn || nv > bn2) { in2 = k; bn2 = nv; firstn = false; }
        }
    }

    float c0x = sr[ip][0],  c0y = sr[ip][1],  c0z = sr[ip][2];
    float c1x = sr[in2][0], c1y = sr[in2][1], c1z = sr[in2][2];
    float mdx = (c1x - c0x) / z, mdy = (c1y - c0y) / z, mdz = (c1z - c0z) / z;
    float ml = sqrtf(mdx * mdx + mdy * mdy + mdz * mdz);
    size_t o = ((size_t)b * SAMP + t) * 3;
    out[o + 0] = c0x + bp * (mdx / ml);
    out[o + 1] = c0y + bp * (mdy / ml);
    out[o + 2] = c0z + bp * (mdz / ml);
}

// ---------------------------------------------------------------------------
extern "C" void kernel_launch(void* const* d_in, const int* in_sizes, int n_in,
                              void* d_out, int out_size, void* d_ws, size_t ws_size,
                              hipStream_t stream) {
    const float* r0     = (const float*)d_in[0];
    const float* m0     = (const float*)d_in[1];
    const float* zv     = (const float*)d_in[2];
    const float* pos    = (const float*)d_in[3];
    const float* nidx   = (const float*)d_in[4];
    const float* origin = (const float*)d_in[5];
    const float* ort    = (const float*)d_in[6];
    const int*   tetra  = (const int*)d_in[7];
    const int*   tface  = (const int*)d_in[8];
    const int*   fvert  = (const int*)d_in[9];
    const int*   ftet   = (const int*)d_in[10];
    float* out = (float*)d_out;

    const int B = in_sizes[0] / 3;
    const int T = in_sizes[5] / 3;
    const int SAMP = in_sizes[2] / B;
    const int NS = 8;

    // workspace carve-up
    char* w = (char*)d_ws;
    size_t off = 0;
    auto carve = [&](size_t bytes) {
        void* p = (void*)(w + off);
        off = (off + bytes + 255) & ~((size_t)255);
        return p;
    };
    float* Aext  = (float*)carve((size_t)4 * T * 4 * sizeof(float));
    float* aob   = (float*)carve((size_t)T * sizeof(float));
    float* nvec  = (float*)carve((size_t)T * 3 * sizeof(float));
    int*   tidx  = (int*)  carve((size_t)B * sizeof(int));
    float* mbuf  = (float*)carve((size_t)B * 3 * sizeof(float));
    float* rhist = (float*)carve((size_t)(NS + 1) * B * 3 * sizeof(float));
    float* dhist = (float*)carve((size_t)(NS + 1) * B * sizeof(float));
    (void)ws_size; (void)n_in; (void)out_size;

    precompute_kernel<<<(T + 63) / 64, 64, 0, stream>>>(pos, tetra, nidx, origin, ort,
                                                        T, Aext, aob, nvec);
    init_kernel<<<(B * 3 + 255) / 256, 256, 0, stream>>>(r0, m0, rhist, mbuf, dhist, B);

    const int findBlocks = (B + 15) / 16;   // 256 threads = 8 waves per 16 rays
    find_tetra_kernel<<<findBlocks, 256, 0, stream>>>(rhist, Aext, tidx, T, B);

    for (int s = 0; s < NS; ++s) {
        evolve_kernel<<<(B + 255) / 256, 256, 0, stream>>>(
            tidx,
            rhist + (size_t)s * B * 3,
            rhist + (size_t)(s + 1) * B * 3,
            mbuf,
            dhist + (size_t)(s + 1) * B,
            aob, nvec, pos, tface, fvert, ftet, T, B);
        if (s < NS - 1) {
            find_tetra_kernel<<<findBlocks, 256, 0, stream>>>(
                rhist + (size_t)(s + 1) * B * 3, Aext, tidx, T, B);
        }
    }

    finalize_kernel<<<B, SAMP, 0, stream>>>(zv, rhist, dhist, out, B, SAMP);
}